// FocalLoss_75033078661163
// MI455X (gfx1250) — compile-verified
//
#include <hip/hip_runtime.h>

// Focal loss: loss_i = -ALPHA * (1 - log(p_i[t_i]))^2 * log(p_i[t_i]); out = sum_i loss_i
// prd: [65536, 4096] f32, trg: [65536] i64, out: scalar f32.

#define FL_ALPHA 0.25f
#define FL_BATCH 65536
#define FL_NCLS  4096
#define BLOCK    256
#define GRID     (FL_BATCH / BLOCK)   // 256 blocks -> 256 partials in d_ws

typedef __attribute__((ext_vector_type(2))) float v2f;
typedef __attribute__((ext_vector_type(8))) float v8f;

// Wave32 sum using V_WMMA_F32_16X16X4_F32 as the first halving step.
// A = ones(16x4), B(vgpr0)=s, B(vgpr1)=0  => every D element of column n equals
// s[n] + s[n+16] (lane L and L+16 share column N with distinct K under the f32
// B layout). D vgpr0 across all 32 lanes holds t[L&15] = s[L&15] + s[(L&15)+16];
// 4 shfl_xor steps then sum the 16 distinct column values.
// Requires EXEC == all ones (call only from uniform control flow).
__device__ __forceinline__ float wave_sum_wmma(float s) {
  v2f a = {1.0f, 1.0f};
  v2f b = {s, 0.0f};
  v8f c = {};
  v8f d = __builtin_amdgcn_wmma_f32_16x16x4_f32(
      /*neg_a=*/false, a, /*neg_b=*/false, b,
      /*c_mod=*/(short)0, c, /*reuse_a=*/false, /*reuse_b=*/false);
  float r = d[0];
  r += __shfl_xor(r, 8, 32);
  r += __shfl_xor(r, 4, 32);
  r += __shfl_xor(r, 2, 32);
  r += __shfl_xor(r, 1, 32);
  return r;  // wave total, replicated in all lanes
}

__global__ __launch_bounds__(BLOCK)
void focal_partial_kernel(const float* __restrict__ prd,
                          const long long* __restrict__ trg,
                          float* __restrict__ partial) {
  __shared__ long long strg[BLOCK];  // async-staged targets
  __shared__ float wsum[BLOCK / 32];

  const unsigned tid = threadIdx.x;
  const unsigned gid = blockIdx.x * BLOCK + tid;

  // Stage this thread's int64 target into LDS via the gfx1250 async
  // global->LDS engine (GVS addressing: saddr base + 32-bit voffset),
  // then wait on ASYNCcnt. Each lane reads back only its own slot, so no
  // cross-lane barrier is needed before the LDS read.
  unsigned lds_off = (unsigned)(size_t)(&strg[tid]);  // low 32 bits = LDS offset
  unsigned voff    = gid * 8u;                        // byte offset into trg
  asm volatile("global_load_async_to_lds_b64 %0, %1, %2\n\t"
               "s_wait_asynccnt 0"
               :: "v"(lds_off), "v"(voff), "s"(trg)
               : "memory");

  const int t = (int)strg[tid];                       // 0 <= t < 4096
  const float p = prd[(size_t)gid * FL_NCLS + (unsigned)t];  // random-column gather
  const float lp = __logf(p);
  const float om = 1.0f - lp;
  const float s = -FL_ALPHA * om * om * lp;

  // Uniform control flow up to here: EXEC is all ones for the WMMA.
  const float r = wave_sum_wmma(s);

  if ((tid & 31u) == 0u) wsum[tid >> 5] = r;
  __syncthreads();
  if (tid == 0) {
    float acc = 0.0f;
#pragma unroll
    for (int i = 0; i < BLOCK / 32; ++i) acc += wsum[i];
    partial[blockIdx.x] = acc;  // deterministic per-block partial
  }
}

__global__ __launch_bounds__(32)
void focal_final_kernel(const float* __restrict__ partial,
                        float* __restrict__ out) {
  // One wave32 reduces GRID (=256) partials in a fixed order.
  float s = 0.0f;
#pragma unroll
  for (int j = 0; j < GRID / 32; ++j) s += partial[threadIdx.x + 32 * j];

  const float r = wave_sum_wmma(s);  // EXEC all ones: 32 threads, uniform flow
  if (threadIdx.x == 0) out[0] = r;
}

extern "C" void kernel_launch(void* const* d_in, const int* in_sizes, int n_in,
                              void* d_out, int out_size, void* d_ws, size_t ws_size,
                              hipStream_t stream) {
  const float*     prd = (const float*)d_in[0];
  const long long* trg = (const long long*)d_in[1];
  float*           ws  = (float*)d_ws;   // GRID floats of scratch (1 KB)
  float*           out = (float*)d_out;

  focal_partial_kernel<<<GRID, BLOCK, 0, stream>>>(prd, trg, ws);
  focal_final_kernel<<<1, 32, 0, stream>>>(ws, out);
}